// GATEncoder_10514079941368
// MI455X (gfx1250) — compile-verified
//
#include <hip/hip_runtime.h>
#include <hip/hip_bf16.h>
#include <math.h>

// ---------------------------------------------------------------------------
// GATv2 encoder for MI455X (gfx1250, wave32).
// GEMMs use V_WMMA_F32_16X16X4_F32 (full-f32 matrix cores, CDNA5 WMMA path).
// Edge phase: wave-per-edge coalesced 1KB-row gathers + f32 atomics; the
// XL/XR/ACC working set (~154MB) fits in the 192MB L2, so random edge traffic
// stays on-chip.
// ---------------------------------------------------------------------------

typedef float v2f __attribute__((ext_vector_type(2)));
typedef float v8f __attribute__((ext_vector_type(8)));

#define NEG_SLOPE 0.2f
#define BN_EPS 1e-5f

__device__ __forceinline__ unsigned fenc(float f) {
  unsigned b = __float_as_uint(f);
  return (b & 0x80000000u) ? ~b : (b | 0x80000000u);
}
__device__ __forceinline__ float fdec(unsigned u) {
  return __uint_as_float((u & 0x80000000u) ? (u & 0x7fffffffu) : ~u);
}
__device__ __forceinline__ float lrelu(float x) {
  return x > 0.0f ? x : NEG_SLOPE * x;
}

// C[M,256] = A[M,K] @ B[K,256], K % 4 == 0. One wave -> one 16x16 C tile.
// Fragment layouts per CDNA5 ISA 7.12.2 (f32 16x16x4).
__global__ void gemm_wmma_f32(const float* __restrict__ A,
                              const float* __restrict__ B,
                              float* __restrict__ C, int M, int K) {
  const int lane = threadIdx.x;              // 0..31
  const int half = lane >> 4;                // 0 or 1
  const int l16  = lane & 15;
  const int m0 = blockIdx.x * 16;
  const int n0 = (blockIdx.y * blockDim.y + threadIdx.y) * 16;
  int mrow = m0 + l16;
  if (mrow >= M) mrow = M - 1;               // safe load; store is guarded
  v8f c = {};
  for (int k0 = 0; k0 < K; k0 += 4) {
    v2f a, b;
    const float* ap = A + (size_t)mrow * K + k0 + half * 2;
    a.x = ap[0];
    a.y = ap[1];
    const float* bp = B + (size_t)(k0 + half * 2) * 256 + n0 + l16;
    b.x = bp[0];
    b.y = bp[256];
    c = __builtin_amdgcn_wmma_f32_16x16x4_f32(false, a, false, b,
                                              (short)0, c, false, false);
  }
#pragma unroll
  for (int v = 0; v < 8; ++v) {
    int row = m0 + half * 8 + v;
    if (row < M) C[(size_t)row * 256 + n0 + l16] = c[v];
  }
}

// One wave per edge: score[e,h] = sum_c att[h,c]*lrelu(xl[src]+xr[dst]),
// plus atomic segment-max per (dst, head) using ordered-uint encoding.
__global__ void edge_score_k(const float* __restrict__ XL,
                             const float* __restrict__ XR,
                             const int* __restrict__ ei, int E, int Etot,
                             const float* __restrict__ att,
                             float* __restrict__ score,
                             unsigned* __restrict__ nmax) {
  int wid = blockIdx.x * (blockDim.x >> 5) + (threadIdx.x >> 5);
  if (wid >= Etot) return;
  int lane = threadIdx.x & 31;
  int s, d;
  if (wid < E) { s = ei[wid]; d = ei[E + wid]; }
  else         { s = wid - E; d = s; }
  const float* xl = XL + (size_t)s * 256;
  const float* xr = XR + (size_t)d * 256;
  int base = lane * 8;                       // 8 elems/lane, one head per 8 lanes
  float4 a0 = *(const float4*)(xl + base);
  float4 a1 = *(const float4*)(xl + base + 4);
  float4 b0 = *(const float4*)(xr + base);
  float4 b1 = *(const float4*)(xr + base + 4);
  float4 t0 = *(const float4*)(att + base);
  float4 t1 = *(const float4*)(att + base + 4);
  float p = t0.x * lrelu(a0.x + b0.x) + t0.y * lrelu(a0.y + b0.y)
          + t0.z * lrelu(a0.z + b0.z) + t0.w * lrelu(a0.w + b0.w)
          + t1.x * lrelu(a1.x + b1.x) + t1.y * lrelu(a1.y + b1.y)
          + t1.z * lrelu(a1.z + b1.z) + t1.w * lrelu(a1.w + b1.w);
  p += __shfl_xor(p, 1, 32);
  p += __shfl_xor(p, 2, 32);
  p += __shfl_xor(p, 4, 32);
  if ((lane & 7) == 0) {
    int h = lane >> 3;
    score[(size_t)wid * 4 + h] = p;
    atomicMax(&nmax[(size_t)d * 4 + h], fenc(p));
  }
}

// One thread per (edge, head): expv = exp(score - max[dst]); accumulate denom.
__global__ void edge_exp_k(float* __restrict__ score,
                           const unsigned* __restrict__ nmax,
                           float* __restrict__ nden,
                           const int* __restrict__ ei, int E, int Etot) {
  int t = blockIdx.x * blockDim.x + threadIdx.x;
  if (t >= Etot * 4) return;
  int e = t >> 2, h = t & 3;
  int d = (e < E) ? ei[E + e] : (e - E);
  float m = fdec(nmax[(size_t)d * 4 + h]);
  float v = expf(score[t] - m);
  score[t] = v;                               // overwrite with expv
  atomicAdd(&nden[(size_t)d * 4 + h], v);
}

// One wave per edge: acc[dst] += xl[src] * alpha  (256 f32 atomics per edge).
__global__ void edge_agg_k(const float* __restrict__ XL,
                           const float* __restrict__ score,
                           const float* __restrict__ nden,
                           const int* __restrict__ ei, int E, int Etot,
                           float* __restrict__ acc) {
  int wid = blockIdx.x * (blockDim.x >> 5) + (threadIdx.x >> 5);
  if (wid >= Etot) return;
  int lane = threadIdx.x & 31;
  int s, d;
  if (wid < E) { s = ei[wid]; d = ei[E + wid]; }
  else         { s = wid - E; d = s; }
  int h = lane >> 3;
  float alpha = score[(size_t)wid * 4 + h] / nden[(size_t)d * 4 + h];
  const float* xl = XL + (size_t)s * 256;
  float* ap = acc + (size_t)d * 256;
  int base = lane * 8;
  float4 a0 = *(const float4*)(xl + base);
  float4 a1 = *(const float4*)(xl + base + 4);
  atomicAdd(ap + base + 0, a0.x * alpha);
  atomicAdd(ap + base + 1, a0.y * alpha);
  atomicAdd(ap + base + 2, a0.z * alpha);
  atomicAdd(ap + base + 3, a0.w * alpha);
  atomicAdd(ap + base + 4, a1.x * alpha);
  atomicAdd(ap + base + 5, a1.y * alpha);
  atomicAdd(ap + base + 6, a1.z * alpha);
  atomicAdd(ap + base + 7, a1.w * alpha);
}

// hb[n,c] = mean over 4 heads of acc[n, h*64+c] + bias[c]
__global__ void head_mean_k(const float* __restrict__ acc,
                            const float* __restrict__ bias, int N,
                            float* __restrict__ hb) {
  int t = blockIdx.x * blockDim.x + threadIdx.x;
  if (t >= N * 64) return;
  int n = t >> 6, c = t & 63;
  const float* a = acc + (size_t)n * 256 + c;
  hb[t] = 0.25f * (a[0] + a[64] + a[128] + a[192]) + bias[c];
}

// Per-channel sum / sum-of-squares over nodes (for batchnorm statistics).
__global__ void col_stats_k(const float* __restrict__ H, int N,
                            float* __restrict__ gsum, float* __restrict__ gsum2) {
  int c = threadIdx.x & 63;
  int local = threadIdx.x >> 6;               // 0..3
  float s = 0.0f, s2 = 0.0f;
  for (int n = blockIdx.x * 4 + local; n < N; n += gridDim.x * 4) {
    float v = H[(size_t)n * 64 + c];
    s += v;
    s2 += v * v;
  }
  __shared__ float ls[256], ls2[256];
  ls[threadIdx.x] = s;
  ls2[threadIdx.x] = s2;
  __syncthreads();
  if (local == 0) {
    atomicAdd(&gsum[c],  ls[c] + ls[64 + c] + ls[128 + c] + ls[192 + c]);
    atomicAdd(&gsum2[c], ls2[c] + ls2[64 + c] + ls2[128 + c] + ls2[192 + c]);
  }
}

// relu(batchnorm(hb)) concat l, zero-padded to 68 cols (layer-2 GEMM input).
__global__ void bn_concat_k(const float* __restrict__ hb,
                            const float* __restrict__ lvec,
                            const float* __restrict__ gamma,
                            const float* __restrict__ beta,
                            const float* __restrict__ gsum,
                            const float* __restrict__ gsum2, int N,
                            float* __restrict__ x2) {
  int t = blockIdx.x * blockDim.x + threadIdx.x;
  if (t >= N * 68) return;
  int n = t / 68, c = t % 68;
  float v;
  if (c < 64) {
    float invN = 1.0f / (float)N;
    float mu  = gsum[c] * invN;
    float var = gsum2[c] * invN - mu * mu;
    float x = (hb[(size_t)n * 64 + c] - mu) * rsqrtf(var + BN_EPS) * gamma[c] + beta[c];
    v = x > 0.0f ? x : 0.0f;
  } else if (c == 64) {
    v = lvec[n];
  } else {
    v = 0.0f;
  }
  x2[t] = v;
}

// Zero-pad [65,256] weights to [68,256] for K%4==0 WMMA GEMM.
__global__ void pad_w_k(const float* __restrict__ Wl, const float* __restrict__ Wr,
                        float* __restrict__ WlP, float* __restrict__ WrP) {
  int t = blockIdx.x * blockDim.x + threadIdx.x;
  const int total = 68 * 256;
  if (t >= 2 * total) return;
  int w = t / total;
  int idx = t % total;
  int r = idx >> 8;
  float v = (r < 65) ? (w ? Wr[idx] : Wl[idx]) : 0.0f;
  (w ? WrP : WlP)[idx] = v;
}

// Column sums of relu(batchnorm(hb)) -> colsum[64] (mean commutes with Wlin).
__global__ void bn_colsum_k(const float* __restrict__ hb,
                            const float* __restrict__ gamma,
                            const float* __restrict__ beta,
                            const float* __restrict__ gsum,
                            const float* __restrict__ gsum2, int N,
                            float* __restrict__ colsum) {
  int c = threadIdx.x & 63;
  int local = threadIdx.x >> 6;
  float invN = 1.0f / (float)N;
  float mu  = gsum[c] * invN;
  float var = gsum2[c] * invN - mu * mu;
  float rs = rsqrtf(var + BN_EPS) * gamma[c];
  float bt = beta[c];
  float s = 0.0f;
  for (int n = blockIdx.x * 4 + local; n < N; n += gridDim.x * 4) {
    float x = (hb[(size_t)n * 64 + c] - mu) * rs + bt;
    s += x > 0.0f ? x : 0.0f;
  }
  __shared__ float ls[256];
  ls[threadIdx.x] = s;
  __syncthreads();
  if (local == 0)
    atomicAdd(&colsum[c], ls[c] + ls[64 + c] + ls[128 + c] + ls[192 + c]);
}

// out[j] = (colsum/N) @ Wlin[:,j] + blin[j]
__global__ void final_linear_k(const float* __restrict__ colsum,
                               const float* __restrict__ Wlin,
                               const float* __restrict__ blin, int N,
                               float* __restrict__ out) {
  int j = threadIdx.x;
  if (j >= 128) return;
  float invN = 1.0f / (float)N;
  float acc = blin[j];
  for (int c = 0; c < 64; ++c)
    acc += (colsum[c] * invN) * Wlin[c * 128 + j];
  out[j] = acc;
}

extern "C" void kernel_launch(void* const* d_in, const int* in_sizes, int n_in,
                              void* d_out, int out_size, void* d_ws, size_t ws_size,
                              hipStream_t stream) {
  const float* x    = (const float*)d_in[0];
  const int*   ei   = (const int*)d_in[1];   // [2,E] (src row, then dst row)
  const float* lvec = (const float*)d_in[2];
  const float* Wl1  = (const float*)d_in[3];
  const float* Wr1  = (const float*)d_in[4];
  const float* att1 = (const float*)d_in[5];
  const float* b1   = (const float*)d_in[6];
  const float* g1   = (const float*)d_in[7];
  const float* be1  = (const float*)d_in[8];
  const float* Wl2  = (const float*)d_in[9];
  const float* Wr2  = (const float*)d_in[10];
  const float* att2 = (const float*)d_in[11];
  const float* b2   = (const float*)d_in[12];
  const float* g2   = (const float*)d_in[13];
  const float* be2  = (const float*)d_in[14];
  const float* Wlin = (const float*)d_in[15];
  const float* blin = (const float*)d_in[16];

  const int N    = in_sizes[0] / 32;
  const int E    = in_sizes[1] / 2;
  const int Etot = E + N;                    // self-loops appended

  // ---- workspace carve-up (256B aligned) ----
  char* ws = (char*)d_ws;
  size_t off = 0;
  auto carve = [&](size_t bytes) -> char* {
    char* p = ws + off;
    off = (off + bytes + 255) & ~(size_t)255;
    return p;
  };
  float*    XL     = (float*)carve((size_t)N * 256 * 4);
  float*    XR     = (float*)carve((size_t)N * 256 * 4);
  float*    ACC    = (float*)carve((size_t)N * 256 * 4);
  float*    SCORE  = (float*)carve((size_t)Etot * 4 * 4);
  unsigned* NMAX   = (unsigned*)carve((size_t)N * 4 * 4);
  float*    NDEN   = (float*)carve((size_t)N * 4 * 4);
  float*    HB     = (float*)carve((size_t)N * 64 * 4);
  float*    X2P    = (float*)carve((size_t)N * 68 * 4);
  float*    W2LP   = (float*)carve((size_t)68 * 256 * 4);
  float*    W2RP   = (float*)carve((size_t)68 * 256 * 4);
  float*    GSUM   = (float*)carve(64 * 4);
  float*    GSUM2  = (float*)carve(64 * 4);
  float*    COLSUM = (float*)carve(64 * 4);

  const int  Mtiles = (N + 15) / 16;
  const dim3 gemmGrid(Mtiles, 4), gemmBlock(32, 4);
  const int  waveBlocks = (Etot + 7) / 8;               // 8 waves / 256-thr block
  const int  ehBlocks   = (Etot * 4 + 255) / 256;

  // ================= Layer 1 =================
  gemm_wmma_f32<<<gemmGrid, gemmBlock, 0, stream>>>(x, Wl1, XL, N, 32);
  gemm_wmma_f32<<<gemmGrid, gemmBlock, 0, stream>>>(x, Wr1, XR, N, 32);

  hipMemsetAsync(ACC,  0, (size_t)N * 256 * 4, stream);
  hipMemsetAsync(NMAX, 0, (size_t)N * 4 * 4,  stream);  // 0 == encoded -NaN (identity)
  hipMemsetAsync(NDEN, 0, (size_t)N * 4 * 4,  stream);
  hipMemsetAsync(GSUM, 0, 64 * 4, stream);
  hipMemsetAsync(GSUM2,0, 64 * 4, stream);

  edge_score_k<<<waveBlocks, 256, 0, stream>>>(XL, XR, ei, E, Etot, att1, SCORE, NMAX);
  edge_exp_k  <<<ehBlocks,  256, 0, stream>>>(SCORE, NMAX, NDEN, ei, E, Etot);
  edge_agg_k  <<<waveBlocks, 256, 0, stream>>>(XL, SCORE, NDEN, ei, E, Etot, ACC);
  head_mean_k <<<(N * 64 + 255) / 256, 256, 0, stream>>>(ACC, b1, N, HB);
  col_stats_k <<<512, 256, 0, stream>>>(HB, N, GSUM, GSUM2);
  bn_concat_k <<<(N * 68 + 255) / 256, 256, 0, stream>>>(HB, lvec, g1, be1, GSUM, GSUM2, N, X2P);

  // ================= Layer 2 =================
  pad_w_k<<<(2 * 68 * 256 + 255) / 256, 256, 0, stream>>>(Wl2, Wr2, W2LP, W2RP);
  gemm_wmma_f32<<<gemmGrid, gemmBlock, 0, stream>>>(X2P, W2LP, XL, N, 68);
  gemm_wmma_f32<<<gemmGrid, gemmBlock, 0, stream>>>(X2P, W2RP, XR, N, 68);

  hipMemsetAsync(ACC,  0, (size_t)N * 256 * 4, stream);
  hipMemsetAsync(NMAX, 0, (size_t)N * 4 * 4,  stream);
  hipMemsetAsync(NDEN, 0, (size_t)N * 4 * 4,  stream);
  hipMemsetAsync(GSUM, 0, 64 * 4, stream);
  hipMemsetAsync(GSUM2,0, 64 * 4, stream);
  hipMemsetAsync(COLSUM, 0, 64 * 4, stream);

  edge_score_k<<<waveBlocks, 256, 0, stream>>>(XL, XR, ei, E, Etot, att2, SCORE, NMAX);
  edge_exp_k  <<<ehBlocks,  256, 0, stream>>>(SCORE, NMAX, NDEN, ei, E, Etot);
  edge_agg_k  <<<waveBlocks, 256, 0, stream>>>(XL, SCORE, NDEN, ei, E, Etot, ACC);
  head_mean_k <<<(N * 64 + 255) / 256, 256, 0, stream>>>(ACC, b2, N, HB);
  col_stats_k <<<512, 256, 0, stream>>>(HB, N, GSUM, GSUM2);
  bn_colsum_k <<<512, 256, 0, stream>>>(HB, g2, be2, GSUM, GSUM2, N, COLSUM);

  final_linear_k<<<1, 128, 0, stream>>>(COLSUM, Wlin, blin, N, (float*)d_out);

  (void)n_in; (void)out_size; (void)ws_size;
}